// ActorCriticNetwork_51668456571053
// MI455X (gfx1250) — compile-verified
//
#include <hip/hip_runtime.h>
#include <hip/hip_bf16.h>
#include <stdint.h>

// ---------------------------------------------------------------------------
// ActorCritic pointer network forward for MI455X (gfx1250, wave32).
// - All GEMM-shaped math: V_WMMA_F32_16X16X4_F32 (fp32 matrix pipe, preserves
//   reference fp32 semantics; K=128 -> 32 WMMA issues per 16x16 tile).
// - GCN edge matrices generated on the fly into the WMMA A operand (the 67MB
//   edges tensor never touches HBM).
// - LSTM recurrence (the serial critical path: 1024 dependent steps total)
//   keeps h/c in LDS and double-buffers the per-step zx slice in LDS via
//   GLOBAL_LOAD_ASYNC_TO_LDS_B128 (ASYNCcnt), prefetching step t+1 during
//   step t's WMMA chain. One A fragment (h slice) feeds 4 independent WMMA
//   accumulators per k-step (4x less LDS traffic, no WMMA->WMMA RAW chain).
// ---------------------------------------------------------------------------

typedef __attribute__((ext_vector_type(2))) float v2f;
typedef __attribute__((ext_vector_type(8))) float v8f;

#define BATCH 256
#define NN    256
#define HH    128
#define G4    512
#define NACT  8

static __device__ __forceinline__ v8f wmma_f32(v2f a, v2f b, v8f c) {
  // D = A(16x4 f32) * B(4x16 f32) + C(16x16 f32)
  return __builtin_amdgcn_wmma_f32_16x16x4_f32(false, a, false, b, (short)0, c, false, false);
}

// async copy 16B global -> LDS, tracked by ASYNCcnt
static __device__ __forceinline__ void async_g2l_b128(void* lds, const void* gptr) {
  asm volatile("global_load_async_to_lds_b128 %0, %1, off"
               :: "v"((unsigned)(uintptr_t)lds), "v"((unsigned long long)(uintptr_t)gptr)
               : "memory");
}
static __device__ __forceinline__ void wait_async0() {
  asm volatile("s_wait_asynccnt 0" ::: "memory");
}

// ---------------------------------------------------------------------------
// Embedding: out[row,h] = W[h,0]*in0 + W[h,1]*in1 + b[h]   (K=2, VALU)
// ---------------------------------------------------------------------------
__global__ void k_embed(const float* __restrict__ inp, const float* __restrict__ W,
                        const float* __restrict__ b, float* __restrict__ out, int total) {
  int idx = blockIdx.x * blockDim.x + threadIdx.x;
  if (idx >= total) return;
  int row = idx >> 7, h = idx & 127;
  float x0 = inp[row * 2], x1 = inp[row * 2 + 1];
  out[idx] = W[h * 2] * x0 + W[h * 2 + 1] * x1 + b[h];
}

// ---------------------------------------------------------------------------
// Generic y = act( X @ W^T + b1 (+b2) (+y) ), one 16x16 tile per wave.
// ---------------------------------------------------------------------------
#define ACT_NONE 0
#define ACT_RELU 1
#define ACT_TANH 2

__global__ void k_lin_wmma(const float* __restrict__ X, long ldx,
                           const float* __restrict__ W,
                           const float* __restrict__ b1, const float* __restrict__ b2,
                           float* __restrict__ out, long ldo,
                           int R, int O, int K, int act, int accum) {
  int wave = (blockIdx.x * blockDim.x + threadIdx.x) >> 5;
  int lane = threadIdx.x & 31;
  int tilesO = O >> 4;
  int nTiles = (R >> 4) * tilesO;
  if (wave >= nTiles) return;
  int tr = wave / tilesO, to = wave % tilesO;
  int m = lane & 15;             // A row / B col within tile
  int kk = (lane >> 4) << 1;     // 0 or 2
  int half = lane >> 4;
  const float* xrow = X + (long)(tr * 16 + m) * ldx;
  const float* wrow = W + (long)(to * 16 + m) * K;
  v8f acc = {};
#pragma unroll 8
  for (int k0 = 0; k0 < K; k0 += 4) {
    v2f a, bb;
    a.x = xrow[k0 + kk];  a.y = xrow[k0 + kk + 1];
    bb.x = wrow[k0 + kk]; bb.y = wrow[k0 + kk + 1];
    acc = wmma_f32(a, bb, acc);
  }
  int n = to * 16 + m;
  float bias = (b1 ? b1[n] : 0.f) + (b2 ? b2[n] : 0.f);
#pragma unroll
  for (int r = 0; r < 8; ++r) {
    int mm = tr * 16 + r + half * 8;
    long off = (long)mm * ldo + n;
    float v = acc[r] + bias;
    if (accum) v += out[off];
    if (act == ACT_RELU) v = fmaxf(v, 0.f);
    else if (act == ACT_TANH) v = tanhf(v);
    out[off] = v;
  }
}

// ---------------------------------------------------------------------------
// Fused graph conv: out[b,i,h] = res[b,i,h] + relu( sum_j edge(b,i,j)*y[b,j,h] )
// ---------------------------------------------------------------------------
__global__ void k_gcn_wmma(const float* __restrict__ inp,  // B,N,2
                           const float* __restrict__ res,  // B,N,H
                           const float* __restrict__ y,    // B,N,H
                           float* __restrict__ out) {
  int wave = (blockIdx.x * blockDim.x + threadIdx.x) >> 5;
  int lane = threadIdx.x & 31;
  const int tilesH = HH >> 4, tilesI = NN >> 4;
  int nTiles = BATCH * tilesI * tilesH;
  if (wave >= nTiles) return;
  int b = wave / (tilesI * tilesH);
  int rem = wave % (tilesI * tilesH);
  int ti = rem / tilesH, th = rem % tilesH;
  int m = lane & 15, kk = (lane >> 4) << 1, half = lane >> 4;
  int i = ti * 16 + m;
  int h = th * 16 + m;
  const float* ib = inp + (long)b * NN * 2;
  const float* yb = y + (long)b * NN * HH;
  float ix = ib[i * 2], iy = ib[i * 2 + 1];
  v8f acc = {};
#pragma unroll 8
  for (int j0 = 0; j0 < NN; j0 += 4) {
    int j = j0 + kk;
    float dx0 = ix - ib[j * 2],       dy0 = iy - ib[j * 2 + 1];
    float dx1 = ix - ib[(j + 1) * 2], dy1 = iy - ib[(j + 1) * 2 + 1];
    v2f a, bb;
    a.x = dx0 * dx0 + dy0 * dy0;
    a.y = dx1 * dx1 + dy1 * dy1;
    bb.x = yb[(long)j * HH + h];
    bb.y = yb[(long)(j + 1) * HH + h];
    acc = wmma_f32(a, bb, acc);
  }
#pragma unroll
  for (int r = 0; r < 8; ++r) {
    int ii = ti * 16 + r + half * 8;
    long off = ((long)b * NN + ii) * HH + h;
    out[off] = res[off] + fmaxf(acc[r], 0.f);
  }
}

// ---------------------------------------------------------------------------
// LSTM scan. One block = 16 batch rows, 8 waves. h/c in LDS; per-step 16x512
// zx slice double-buffered in LDS via async global->LDS loads (ASYNCcnt).
// Each wave owns a 16x64 slab of z: 4 accumulators share one A fragment per
// k-step. z accumulates in place into the consumed zx buffer.
// Dynamic LDS: (2048 + 2048 + 2*8192) floats = 80KB (of 320KB WGP budget).
// ---------------------------------------------------------------------------
#define LSTM_LDS_BYTES ((2048 + 2048 + 2 * 8192) * 4)

__global__ void k_lstm_scan(const float* __restrict__ zx, long zxB, long zxT,
                            const float* __restrict__ Whh,
                            const float* __restrict__ h_init,
                            const float* __restrict__ c_init,
                            float* __restrict__ s_out, long soN,
                            float* __restrict__ h_fin, float* __restrict__ c_fin,
                            int addFinal, int T, int reverse) {
  extern __shared__ float smem[];
  float* hs  = smem;                 // 16 x 128
  float* cs  = smem + 2048;          // 16 x 128
  float* zb0 = smem + 4096;          // 16 x 512
  float* zb1 = smem + 4096 + 8192;   // 16 x 512
  int b0 = blockIdx.x * 16;
  int tid = threadIdx.x;
  for (int idx = tid; idx < 16 * 128; idx += 256) {
    int bb = idx >> 7, h = idx & 127;
    hs[idx] = h_init ? h_init[(long)(b0 + bb) * 128 + h] : 0.f;
    cs[idx] = c_init ? c_init[(long)(b0 + bb) * 128 + h] : 0.f;
  }
  // async prefetch of step 0's zx slice (2048 x b128, 8 per thread)
  {
    long tm0 = reverse ? (T - 1) : 0;
#pragma unroll
    for (int it = 0; it < 8; ++it) {
      int e4 = tid + it * 256;
      int row = e4 >> 7, col = (e4 & 127) << 2;
      async_g2l_b128(zb0 + row * 512 + col,
                     zx + (long)(b0 + row) * zxB + tm0 * zxT + col);
    }
  }
  wait_async0();
  __syncthreads();

  int wave = tid >> 5, lane = tid & 31;
  int m = lane & 15, kk = (lane >> 4) << 1, half = lane >> 4;
  const float* wr0 = Whh + (long)(wave * 64 + m) * 128;        // n-subtile 0
  const float* wr1 = wr0 + 16 * 128;
  const float* wr2 = wr0 + 32 * 128;
  const float* wr3 = wr0 + 48 * 128;
  for (int t = 0; t < T; ++t) {
    float* bufC = (t & 1) ? zb1 : zb0;
    float* bufN = (t & 1) ? zb0 : zb1;
    // kick off async prefetch of step t+1 before the WMMA chain
    if (t + 1 < T) {
      long tmn = reverse ? (long)(T - 2 - t) : (long)(t + 1);
#pragma unroll
      for (int it = 0; it < 8; ++it) {
        int e4 = tid + it * 256;
        int row = e4 >> 7, col = (e4 & 127) << 2;
        async_g2l_b128(bufN + row * 512 + col,
                       zx + (long)(b0 + row) * zxB + tmn * zxT + col);
      }
    }
    // z = h @ Whh^T + zx : one A fragment feeds 4 independent accumulators
    {
      v8f a0 = {}, a1 = {}, a2 = {}, a3 = {};
#pragma unroll 8
      for (int k0 = 0; k0 < 128; k0 += 4) {
        v2f a;
        a.x = hs[m * 128 + k0 + kk];
        a.y = hs[m * 128 + k0 + kk + 1];
        v2f q0; q0.x = wr0[k0 + kk]; q0.y = wr0[k0 + kk + 1];
        a0 = wmma_f32(a, q0, a0);
        v2f q1; q1.x = wr1[k0 + kk]; q1.y = wr1[k0 + kk + 1];
        a1 = wmma_f32(a, q1, a1);
        v2f q2; q2.x = wr2[k0 + kk]; q2.y = wr2[k0 + kk + 1];
        a2 = wmma_f32(a, q2, a2);
        v2f q3; q3.x = wr3[k0 + kk]; q3.y = wr3[k0 + kk + 1];
        a3 = wmma_f32(a, q3, a3);
      }
      int nb = wave * 64 + m;
#pragma unroll
      for (int r = 0; r < 8; ++r) {
        int mm = r + half * 8;
        float* zrow = bufC + mm * 512 + nb;
        zrow[0]  += a0[r];
        zrow[16] += a1[r];
        zrow[32] += a2[r];
        zrow[48] += a3[r];
      }
    }
    __syncthreads();
    // gates
    long tmem = reverse ? (long)(T - 1 - t) : (long)t;
    for (int idx = tid; idx < 16 * 128; idx += 256) {
      int bb = idx >> 7, h = idx & 127;
      float zi = bufC[bb * 512 + h];
      float zf = bufC[bb * 512 + 128 + h];
      float zg = bufC[bb * 512 + 256 + h];
      float zo = bufC[bb * 512 + 384 + h];
      float si = 1.f / (1.f + __expf(-zi));
      float sf = 1.f / (1.f + __expf(-zf));
      float so = 1.f / (1.f + __expf(-zo));
      float c = sf * cs[idx] + si * tanhf(zg);
      float hh = so * tanhf(c);
      cs[idx] = c;
      hs[idx] = hh;
      if (s_out) s_out[((long)(b0 + bb) * soN + tmem) * 128 + h] = hh;
    }
    wait_async0();   // next step's zx slice resident before anyone proceeds
    __syncthreads();
  }
  for (int idx = tid; idx < 16 * 128; idx += 256) {
    int bb = idx >> 7, h = idx & 127;
    long off = (long)(b0 + bb) * 128 + h;
    if (h_fin) { if (addFinal) h_fin[off] += hs[idx]; else h_fin[off] = hs[idx]; }
    if (c_fin) { if (addFinal) c_fin[off] += cs[idx]; else c_fin[off] = cs[idx]; }
  }
}

// ---------------------------------------------------------------------------
// Decoder helpers
// ---------------------------------------------------------------------------
__global__ void k_dec_init(float* __restrict__ mask, float* __restrict__ dec_input,
                           const float* __restrict__ init_dec,
                           const int* __restrict__ actions, float* __restrict__ ptr_out) {
  int t = blockIdx.x * blockDim.x + threadIdx.x;
  if (t >= BATCH * NN) return;
  int b = t / NN, n = t % NN;
  mask[t] = (n == 0 || n == NN - 1) ? 0.f : 1.f;
  if (n < HH) dec_input[b * HH + n] = init_dec[n];
  if (n < NACT) ptr_out[b * NACT + n] = (float)actions[b * NACT + n];
}

// u_raw[b,n] = sum_g V[g] * tanh(refW1[b,n,g] + hw2[b,g])
__global__ void k_att_u(const float* __restrict__ refW1, const float* __restrict__ hw2,
                        const float* __restrict__ attV, float* __restrict__ u_raw) {
  int t = blockIdx.x * blockDim.x + threadIdx.x;
  if (t >= BATCH * NN) return;
  int b = t / NN;
  const float* rw = refW1 + (long)t * HH;
  const float* hb = hw2 + (long)b * HH;
  float s = 0.f;
#pragma unroll 4
  for (int g = 0; g < HH; ++g) s += attV[g] * tanhf(rw[g] + hb[g]);
  u_raw[t] = s;
}

// One block per batch row: masked clipped softmax, log-prob, entropy,
// mask update, next decoder input gather.
__global__ void k_dec_step(const float* __restrict__ u_raw, float* __restrict__ mask,
                           const int* __restrict__ actions, const float* __restrict__ ref,
                           float* __restrict__ dec_input,
                           float* __restrict__ probs_out, float* __restrict__ logp_out,
                           float* __restrict__ ent_out, int step) {
  __shared__ float red[NN];
  int b = blockIdx.x, n = threadIdx.x;
  const float EPS = 1.1920929e-07f;
  int idx = actions[b * NACT + step];
  float mk = mask[b * NN + n];
  if (step == 0 && n == NN - 2) mk = 0.f;
  if (step == 1 && n == NN - 2) mk = 1.f;
  float u = (mk == 0.f) ? -10.f : 10.f * tanhf(u_raw[b * NN + n]);
  red[n] = u; __syncthreads();
  for (int s2 = NN / 2; s2 >= 1; s2 >>= 1) { if (n < s2) red[n] = fmaxf(red[n], red[n + s2]); __syncthreads(); }
  float mx = red[0]; __syncthreads();
  float e = __expf(u - mx);
  red[n] = e; __syncthreads();
  for (int s2 = NN / 2; s2 >= 1; s2 >>= 1) { if (n < s2) red[n] += red[n + s2]; __syncthreads(); }
  float Z = red[0]; __syncthreads();
  float prob = e / Z;
  probs_out[((long)b * NACT + step) * NN + n] = prob;
  float mp = prob * mk;
  red[n] = mp; __syncthreads();
  for (int s2 = NN / 2; s2 >= 1; s2 >>= 1) { if (n < s2) red[n] += red[n + s2]; __syncthreads(); }
  float S = red[0]; __syncthreads();
  float pn = mp / S;
  if (n == idx) logp_out[b * NACT + step] = __logf(fmaxf(pn, EPS));
  red[n] = pn * __logf(fmaxf(pn, EPS)); __syncthreads();
  for (int s2 = NN / 2; s2 >= 1; s2 >>= 1) { if (n < s2) red[n] += red[n + s2]; __syncthreads(); }
  if (n == 0) ent_out[b * NACT + step] = -red[0];
  mask[b * NN + n] = mk * ((n > idx) ? 1.f : 0.f);
  if (n < HH) dec_input[b * HH + n] = ref[((long)b * NN + idx) * HH + n];
}

// critic head: v[b] = w . x[b,:] + bias
__global__ void k_vhead(const float* __restrict__ x, const float* __restrict__ w,
                        const float* __restrict__ b, float* __restrict__ v) {
  int bi = blockIdx.x * blockDim.x + threadIdx.x;
  if (bi >= BATCH) return;
  float s = b[0];
#pragma unroll 4
  for (int h = 0; h < HH; ++h) s += w[h] * x[bi * HH + h];
  v[bi] = s;
}

// ---------------------------------------------------------------------------
// Host side
// ---------------------------------------------------------------------------
struct EncP {
  const float *Wb_W, *Wb_b, *Wf_W, *Wf_b, *emb_W, *emb_b;
  const float *g0W, *g0b, *g1W, *g1b, *g2W, *g2b;
  const float *rnn_Whh, *rnn_Wih, *rnn_bhh, *rnn_bih;
  const float *rnn0_Whh, *rnn0_Wih, *rnn0_bhh, *rnn0_bih;
  const float *rnn0r_Whh, *rnn0r_Wih, *rnn0r_bhh, *rnn0r_bih;
  const float *rnnr_Whh, *rnnr_Wih, *rnnr_bhh, *rnnr_bih;
};

static EncP loadEnc(void* const* d_in, int base) {
  auto F = [&](int i) { return (const float*)d_in[base + i]; };
  EncP e;
  e.Wb_W = F(0);  e.Wb_b = F(1);  e.Wf_W = F(2);  e.Wf_b = F(3);
  e.emb_W = F(4); e.emb_b = F(5);
  e.g0W = F(6);  e.g0b = F(7);  e.g1W = F(8);  e.g1b = F(9);  e.g2W = F(10); e.g2b = F(11);
  e.rnn_Whh = F(12);  e.rnn_Wih = F(13);  e.rnn_bhh = F(14);  e.rnn_bih = F(15);
  e.rnn0_Whh = F(16); e.rnn0_Wih = F(17); e.rnn0_bhh = F(18); e.rnn0_bih = F(19);
  e.rnn0r_Whh = F(20); e.rnn0r_Wih = F(21); e.rnn0r_bhh = F(22); e.rnn0r_bih = F(23);
  e.rnnr_Whh = F(24); e.rnnr_Wih = F(25); e.rnnr_bhh = F(26); e.rnnr_bih = F(27);
  return e;
}

static inline void linL(const float* X, long ldx, const float* W, const float* b1,
                        const float* b2, float* out, long ldo, int R, int O, int K,
                        int act, int accum, hipStream_t s) {
  int waves = (R >> 4) * (O >> 4);
  int blocks = (waves + 7) / 8;  // 8 waves / 256-thread block
  k_lin_wmma<<<blocks, 256, 0, s>>>(X, ldx, W, b1, b2, out, ldo, R, O, K, act, accum);
}

// ---- workspace layout (floats) ----
#define SZ_BNH ((size_t)BATCH * NN * HH)              // 8,388,608
#define SZ_ZX  ((size_t)BATCH * NN * G4)              // 33,554,432
static const size_t OFF_G     = 0;
static const size_t OFF_TMP   = OFF_G + SZ_BNH;
static const size_t OFF_ZX    = OFF_TMP + SZ_BNH;
static const size_t OFF_SOUTF = OFF_ZX + SZ_ZX;
static const size_t OFF_SOUTR = OFF_SOUTF + SZ_BNH;
static const size_t OFF_REF   = OFF_SOUTR + SZ_BNH;
static const size_t OFF_REFW1 = OFF_REF + SZ_BNH;
static const size_t OFF_Z0    = OFF_REFW1 + SZ_BNH;           // B*512
static const size_t OFF_H0    = OFF_Z0 + (size_t)BATCH * G4;
static const size_t OFF_C0    = OFF_H0 + (size_t)BATCH * HH;
static const size_t OFF_HS    = OFF_C0 + (size_t)BATCH * HH;
static const size_t OFF_CS    = OFF_HS + (size_t)BATCH * HH;
static const size_t OFF_HSTAR = OFF_CS + (size_t)BATCH * HH;
static const size_t OFF_CSTAR = OFF_HSTAR + (size_t)BATCH * HH;
static const size_t OFF_DECHA = OFF_CSTAR + (size_t)BATCH * HH;
static const size_t OFF_DECHB = OFF_DECHA + (size_t)BATCH * HH;
static const size_t OFF_DECIN = OFF_DECHB + (size_t)BATCH * HH;
static const size_t OFF_HW2   = OFF_DECIN + (size_t)BATCH * HH;
static const size_t OFF_URAW  = OFF_HW2 + (size_t)BATCH * HH;
static const size_t OFF_MASK  = OFF_URAW + (size_t)BATCH * NN;
static const size_t OFF_CBUF  = OFF_MASK + (size_t)BATCH * NN;
static const size_t OFF_CTMP  = OFF_CBUF + (size_t)BATCH * HH;
static const size_t WS_FLOATS = OFF_CTMP + (size_t)BATCH * HH;

// run one encoder; writes h_fin/c_fin; if ref_out != nullptr, writes bi-combined s_out.
static void run_encoder(const EncP& e, const float* inp, float* ws,
                        float* h_fin, float* c_fin, float* ref_out, hipStream_t s) {
  float* g    = ws + OFF_G;
  float* tmp  = ws + OFF_TMP;
  float* zx   = ws + OFF_ZX;
  float* z0   = ws + OFF_Z0;
  float* h0   = ws + OFF_H0;
  float* c0   = ws + OFF_C0;
  float* sF   = ws + OFF_SOUTF;
  float* sR   = ws + OFF_SOUTR;
  const int RN = BATCH * NN;

  // embedding
  k_embed<<<(RN * HH + 255) / 256, 256, 0, s>>>(inp, e.emb_W, e.emb_b, g, RN * HH);

  // 3 graph-conv layers: tmp = lin(g); g = g + relu(edges @ tmp)
  const float* gw[3] = {e.g0W, e.g1W, e.g2W};
  const float* gb[3] = {e.g0b, e.g1b, e.g2b};
  int gcnWaves = BATCH * (NN >> 4) * (HH >> 4);
  for (int l = 0; l < 3; ++l) {
    linL(g, HH, gw[l], gb[l], nullptr, tmp, HH, RN, HH, HH, ACT_NONE, 0, s);
    k_gcn_wmma<<<(gcnWaves + 7) / 8, 256, 0, s>>>(inp, g, tmp, g);
  }

  // ---- forward LSTM ----
  linL(g + (size_t)(NN - 1) * HH, (long)NN * HH, e.rnn0_Wih, e.rnn0_bih, e.rnn0_bhh,
       z0, G4, BATCH, G4, HH, ACT_NONE, 0, s);
  k_lstm_scan<<<BATCH / 16, 256, LSTM_LDS_BYTES, s>>>(z0, (long)G4, 0L, e.rnn0_Whh,
      nullptr, nullptr, nullptr, (long)NN, h0, c0, 0, 1, 0);
  linL(g, HH, e.rnn_Wih, e.rnn_bih, e.rnn_bhh, zx, G4, RN, G4, HH, ACT_NONE, 0, s);
  k_lstm_scan<<<BATCH / 16, 256, LSTM_LDS_BYTES, s>>>(zx, (long)NN * G4, (long)G4,
      e.rnn_Whh, h0, c0, sF, (long)NN, h_fin, c_fin, 0, NN, 0);

  // ---- reverse LSTM ---- (gr[:, -1, :] == g[:, 0, :])
  linL(g, (long)NN * HH, e.rnn0r_Wih, e.rnn0r_bih, e.rnn0r_bhh, z0, G4, BATCH, G4, HH,
       ACT_NONE, 0, s);
  k_lstm_scan<<<BATCH / 16, 256, LSTM_LDS_BYTES, s>>>(z0, (long)G4, 0L, e.rnn0r_Whh,
      nullptr, nullptr, nullptr, (long)NN, h0, c0, 0, 1, 0);
  linL(g, HH, e.rnnr_Wih, e.rnnr_bih, e.rnnr_bhh, zx, G4, RN, G4, HH, ACT_NONE, 0, s);
  k_lstm_scan<<<BATCH / 16, 256, LSTM_LDS_BYTES, s>>>(zx, (long)NN * G4, (long)G4,
      e.rnnr_Whh, h0, c0, sR, (long)NN, h_fin, c_fin, 1, NN, 1);

  if (ref_out) {
    // ref = tanh(W_f @ sF + b_f + W_b @ flip(sR) + b_b)
    linL(sF, HH, e.Wf_W, e.Wf_b, nullptr, ref_out, HH, RN, HH, HH, ACT_NONE, 0, s);
    linL(sR, HH, e.Wb_W, e.Wb_b, nullptr, ref_out, HH, RN, HH, HH, ACT_TANH, 1, s);
  }
}

extern "C" void kernel_launch(void* const* d_in, const int* in_sizes, int n_in,
                              void* d_out, int out_size, void* d_ws, size_t ws_size,
                              hipStream_t stream) {
  if (ws_size < WS_FLOATS * sizeof(float)) return;
  float* ws = (float*)d_ws;
  const float* inputs      = (const float*)d_in[0];
  const float* inputs_star = (const float*)d_in[1];
  const int*   actions     = (const int*)d_in[2];
  // params (sorted jax-tree order): W_s, W_star, critic1, critic2, dec, enc, enc_star
  const float* Ws_W    = (const float*)d_in[3];
  const float* Ws_b    = (const float*)d_in[4];
  const float* Wstar_W = (const float*)d_in[5];
  const float* Wstar_b = (const float*)d_in[6];
  const float* c1_W    = (const float*)d_in[7];
  const float* c1_b    = (const float*)d_in[8];
  const float* c2_W    = (const float*)d_in[9];
  const float* c2_b    = (const float*)d_in[10];
  // dec: W_0, W_1, W_s, W_star, att_V, att_W1, att_W2, init_dec
  const float* dW0_W   = (const float*)d_in[11];
  const float* dW0_b   = (const float*)d_in[12];
  const float* dW1_W   = (const float*)d_in[13];
  const float* dW1_b   = (const float*)d_in[14];
  const float* dWs_W   = (const float*)d_in[15];
  const float* dWs_b   = (const float*)d_in[16];
  const float* dWst_W  = (const float*)d_in[17];
  const float* dWst_b  = (const float*)d_in[18];
  const float* attV    = (const float*)d_in[19];
  const float* attW1   = (const float*)d_in[20];
  const float* attW2   = (const float*)d_in[21];
  const float* initDec = (const float*)d_in[22];
  EncP enc      = loadEnc(d_in, 23);
  EncP enc_star = loadEnc(d_in, 51);

  float* hs    = ws + OFF_HS;
  float* cs    = ws + OFF_CS;
  float* hstar = ws + OFF_HSTAR;
  float* cstar = ws + OFF_CSTAR;
  float* ref   = ws + OFF_REF;

  // ---- encoders (enc_star first; its big scratch is then reused by enc) ----
  run_encoder(enc_star, inputs_star, ws, hstar, cstar, nullptr, stream);
  run_encoder(enc, inputs, ws, hs, cs, ref, stream);

  // ---- output layout ----
  float* out = (float*)d_out;
  float* o_probs = out;                          // B * NACT * NN
  float* o_ptr   = out + (size_t)BATCH * NACT * NN;
  float* o_logp  = o_ptr + (size_t)BATCH * NACT;
  float* o_ent   = o_logp + (size_t)BATCH * NACT;
  float* o_v     = o_ent + (size_t)BATCH * NACT;

  // ---- decoder ----
  float* dechA = ws + OFF_DECHA;
  float* dechB = ws + OFF_DECHB;
  float* decin = ws + OFF_DECIN;
  float* hw2   = ws + OFF_HW2;
  float* uraw  = ws + OFF_URAW;
  float* mask  = ws + OFF_MASK;
  float* refW1 = ws + OFF_REFW1;

  // h = concat(lin(dec.W_star, hstar), lin(dec.W_s, hs))
  linL(hstar, HH, dWst_W, dWst_b, nullptr, dechA, HH, BATCH, HH / 2, HH, ACT_NONE, 0, stream);
  linL(hs,    HH, dWs_W,  dWs_b,  nullptr, dechA + HH / 2, HH, BATCH, HH / 2, HH, ACT_NONE, 0, stream);
  // ref_W1 = ref @ att_W1^T
  linL(ref, HH, attW1, nullptr, nullptr, refW1, HH, BATCH * NN, HH, HH, ACT_NONE, 0, stream);
  k_dec_init<<<(BATCH * NN + 255) / 256, 256, 0, stream>>>(mask, decin, initDec, actions, o_ptr);

  float* hcur = dechA;
  float* hnxt = dechB;
  for (int step = 0; step < NACT; ++step) {
    linL(hcur,  HH, dW1_W, dW1_b, nullptr, hnxt, HH, BATCH, HH, HH, ACT_NONE, 0, stream);
    linL(decin, HH, dW0_W, dW0_b, nullptr, hnxt, HH, BATCH, HH, HH, ACT_TANH, 1, stream);
    linL(hnxt, HH, attW2, nullptr, nullptr, hw2, HH, BATCH, HH, HH, ACT_NONE, 0, stream);
    k_att_u<<<(BATCH * NN + 255) / 256, 256, 0, stream>>>(refW1, hw2, attV, uraw);
    k_dec_step<<<BATCH, NN, 0, stream>>>(uraw, mask, actions, ref, decin,
                                         o_probs, o_logp, o_ent, step);
    float* t = hcur; hcur = hnxt; hnxt = t;
  }

  // ---- critic ----
  float* cbuf = ws + OFF_CBUF;
  float* ctmp = ws + OFF_CTMP;
  linL(hstar, HH, Wstar_W, Wstar_b, nullptr, cbuf, HH, BATCH, HH / 2, HH, ACT_NONE, 0, stream);
  linL(hs,    HH, Ws_W,    Ws_b,    nullptr, cbuf + HH / 2, HH, BATCH, HH / 2, HH, ACT_NONE, 0, stream);
  linL(cbuf, HH, c1_W, c1_b, nullptr, ctmp, HH, BATCH, HH, HH, ACT_RELU, 0, stream);
  k_vhead<<<1, BATCH, 0, stream>>>(ctmp, c2_W, c2_b, o_v);
}